// ProposalGenerator_84945863180485
// MI455X (gfx1250) — compile-verified
//
#include <hip/hip_runtime.h>
#include <math.h>

typedef __attribute__((ext_vector_type(2))) float v2f;
typedef __attribute__((ext_vector_type(8))) float v8f;

#define N_ALL     36864   // 64*64*9
#define HW        64
#define PRE_NMS   6000
#define POST_NMS  2000
#define NBINS     2048
#define SEL_CAP   8192
#define CWORDS16  376     // ceil(6000/16) = 375, padded to even -> 188 u32 words
#define CWORDS32  188
#define NMS_T     0.7f

// 9 base anchors (precomputed exactly as _generate_anchors(16) does)
__constant__ float c_anch[9][4] = {
    {-84.f, -40.f,  99.f,  55.f},
    {-176.f,-88.f, 191.f, 103.f},
    {-360.f,-184.f,375.f, 199.f},
    {-56.f, -56.f,  71.f,  71.f},
    {-120.f,-120.f,135.f, 135.f},
    {-248.f,-248.f,263.f, 263.f},
    {-36.f, -84.f,  51.f,  99.f},
    {-80.f, -176.f, 95.f, 191.f},
    {-168.f,-360.f,183.f, 375.f}};

// ---------------- Kernel A: decode + filter ----------------
__global__ __launch_bounds__(256) void decode_kernel(
    const float* __restrict__ deltas, const float* __restrict__ scores,
    const int* __restrict__ imw_p, const int* __restrict__ imh_p,
    float* __restrict__ props, float* __restrict__ sc)
{
    int i = blockIdx.x * 256 + threadIdx.x;
    if (i >= N_ALL) return;
    int a   = i % 9;
    int pix = i / 9;
    int x   = pix & (HW - 1);
    int y   = pix >> 6;
    int off = y * HW + x;

    float d0 = deltas[(a * 4 + 0) * (HW * HW) + off];
    float d1 = deltas[(a * 4 + 1) * (HW * HW) + off];
    float d2 = deltas[(a * 4 + 2) * (HW * HW) + off];
    float d3 = deltas[(a * 4 + 3) * (HW * HW) + off];
    float s  = scores[(9 + a) * (HW * HW) + off];

    float sx = (float)(x * 16), sy = (float)(y * 16);
    float ax1 = c_anch[a][0] + sx, ay1 = c_anch[a][1] + sy;
    float ax2 = c_anch[a][2] + sx, ay2 = c_anch[a][3] + sy;

    float aw = ax2 - ax1 + 1.f, ah = ay2 - ay1 + 1.f;
    float acx = ax1 + 0.5f * aw, acy = ay1 + 0.5f * ah;

    float pcx = d0 * aw + acx, pcy = d1 * ah + acy;
    float pw  = expf(d2) * aw, ph  = expf(d3) * ah;

    float fw = (float)imw_p[0], fh = (float)imh_p[0];
    float x1 = fminf(fmaxf(pcx - 0.5f * pw, 0.f), fw - 1.f);
    float y1 = fminf(fmaxf(pcy - 0.5f * ph, 0.f), fh - 1.f);
    float x2 = fminf(fmaxf(pcx + 0.5f * pw, 0.f), fw - 1.f);
    float y2 = fminf(fmaxf(pcy + 0.5f * ph, 0.f), fh - 1.f);

    float4* pp = (float4*)props;
    pp[i] = make_float4(x1, y1, x2, y2);

    float thresh = 16.f * (fh / fw);
    bool keep = (x2 - x1 + 1.f >= thresh) && (y2 - y1 + 1.f >= thresh);
    sc[i] = keep ? s : -__builtin_inff();
}

// ---------------- Kernel B: score histogram ----------------
__global__ __launch_bounds__(256) void hist_kernel(const float* __restrict__ sc,
                                                   int* __restrict__ hist)
{
    int i = blockIdx.x * 256 + threadIdx.x;
    if (i >= N_ALL) return;
    float s = sc[i];
    if (s >= 0.f) {  // valid scores are uniform [0,1); filtered are -inf
        int b = (int)(s * (float)NBINS);
        b = b < 0 ? 0 : (b > NBINS - 1 ? NBINS - 1 : b);
        atomicAdd(&hist[b], 1);
    }
}

// ---------------- Kernel C: find threshold bin ----------------
__global__ void scan_kernel(const int* __restrict__ hist, int* __restrict__ meta)
{
    int acc = 0, t = 0;
    for (int b = NBINS - 1; b >= 0; --b) {
        acc += hist[b];
        if (acc >= PRE_NMS) { t = b; break; }
    }
    meta[1] = t;
}

// ---------------- Kernel D: compact candidates >= threshold ----------------
__global__ __launch_bounds__(256) void select_kernel(
    const float* __restrict__ sc, int* __restrict__ meta,
    float* __restrict__ selScore, int* __restrict__ selIdx)
{
    int i = blockIdx.x * 256 + threadIdx.x;
    if (i >= N_ALL) return;
    float s = sc[i];
    if (s < 0.f) return;
    int b = (int)(s * (float)NBINS);
    b = b < 0 ? 0 : (b > NBINS - 1 ? NBINS - 1 : b);
    if (b >= meta[1]) {
        int pos = atomicAdd(&meta[0], 1);
        if (pos < SEL_CAP) { selScore[pos] = s; selIdx[pos] = i; }
    }
}

// ---------------- Kernel E: single-block bitonic sort (descending) ----------------
__global__ __launch_bounds__(1024) void sort_kernel(
    const int* __restrict__ meta, const float* __restrict__ selScore,
    const int* __restrict__ selIdx, const float* __restrict__ props,
    float* __restrict__ top_p, float* __restrict__ top_sc)
{
    __shared__ float key[SEL_CAP];
    __shared__ int   val[SEL_CAP];
    int m = meta[0]; if (m > SEL_CAP) m = SEL_CAP;

    for (int i = threadIdx.x; i < SEL_CAP; i += 1024) {
        if (i < m) { key[i] = selScore[i]; val[i] = selIdx[i]; }
        else       { key[i] = -__builtin_inff(); val[i] = 0; }
    }
    __syncthreads();

    for (int k = 2; k <= SEL_CAP; k <<= 1) {
        for (int j = k >> 1; j > 0; j >>= 1) {
            for (int i = threadIdx.x; i < SEL_CAP; i += 1024) {
                int ixj = i ^ j;
                if (ixj > i) {
                    bool up = ((i & k) == 0);   // descending overall
                    float ki = key[i], kj = key[ixj];
                    bool sw = up ? (ki < kj) : (ki > kj);
                    if (sw) {
                        key[i] = kj; key[ixj] = ki;
                        int t = val[i]; val[i] = val[ixj]; val[ixj] = t;
                    }
                }
            }
            __syncthreads();
        }
    }

    const float4* pp = (const float4*)props;
    float4* tp = (float4*)top_p;
    for (int i = threadIdx.x; i < PRE_NMS; i += 1024) {
        top_sc[i] = key[i];
        tp[i] = pp[val[i]];
    }
}

// ---------------- Kernel F: suppression bitmask tiles (uses WMMA) ----------------
// One wave32 per 16x16 tile. D = area_row + area_col computed by
// V_WMMA_F32_16X16X4_F32 as a rank-2 outer product; min/max intersection in VALU
// co-executes with the matrix pipe. Ballot packs 16+16 suppression bits per row.
__global__ __launch_bounds__(32) void sup_kernel(
    const float* __restrict__ top_p, unsigned short* __restrict__ sup16)
{
    int lane = threadIdx.x;
    int rowbase = blockIdx.y * 16;
    int colbase = blockIdx.x * 16;
    int half = lane >> 4;
    int ln   = lane & 15;
    int col  = colbase + ln;

    const float4* tp = (const float4*)top_p;
    float4 cb = tp[col];
    float carea = (cb.z - cb.x + 1.f) * (cb.w - cb.y + 1.f);

    // A (16x4): col0 = area of rows, col1 = ones, K=2,3 zero (lanes 16-31).
    // B (4x16): row0 = ones, row1 = area of cols, K=2,3 zero.
    v2f A = {0.f, 0.f};
    v2f B = {0.f, 0.f};
    if (half == 0) {
        float4 rb = tp[rowbase + ln];
        A.x = (rb.z - rb.x + 1.f) * (rb.w - rb.y + 1.f);
        A.y = 1.f;
        B.x = 1.f;
        B.y = carea;
    }
    v8f C = {};
    // D[v] (lane l) = area_row[rowbase + v + 8*(l>=16)] + area_col[colbase + l%16]
    v8f D = __builtin_amdgcn_wmma_f32_16x16x4_f32(
        false, A, false, B, (short)0, C, false, false);

    const float T = NMS_T, c1 = 1.f + NMS_T;
#pragma unroll
    for (int r = 0; r < 8; ++r) {
        int row = rowbase + half * 8 + r;
        float4 rb = tp[row];
        float iw = fminf(cb.z, rb.z) - fmaxf(cb.x, rb.x) + 1.f;
        float ih = fminf(cb.w, rb.w) - fmaxf(cb.y, rb.y) + 1.f;
        iw = fmaxf(iw, 0.f); ih = fmaxf(ih, 0.f);
        float inter = iw * ih;
        // iou > T  <=>  inter*(1+T) > T*(area_i + area_j)
        bool pred = (col > row) && (inter * c1 > T * D[r]);
        unsigned int m = __builtin_amdgcn_ballot_w32(pred);
        if (lane == 0) {
            sup16[(size_t)(rowbase + r)     * CWORDS16 + blockIdx.x] = (unsigned short)(m & 0xFFFFu);
            sup16[(size_t)(rowbase + 8 + r) * CWORDS16 + blockIdx.x] = (unsigned short)(m >> 16);
        }
    }
}

// ---------------- async staging helper (CDNA5 async global->LDS DMA path) ----------
// Fetch one 752-byte suppression row into an LDS staging buffer using
// GLOBAL_LOAD_ASYNC_TO_LDS_B128 (tracked by ASYNCcnt): 32 lanes x 16B + 15 lanes x 16B.
__device__ __forceinline__ void async_fetch_row(
    const unsigned short* sup16, int row,
    unsigned int (*stage)[CWORDS32], int buf, int lane)
{
#if defined(__HIP_DEVICE_COMPILE__)
    const unsigned int* rowp = (const unsigned int*)(sup16 + (size_t)row * CWORDS16);
    unsigned int       lds0 = (unsigned int)(size_t)&stage[buf][lane * 4];
    unsigned long long ga0  = (unsigned long long)(size_t)(rowp + lane * 4);
    asm volatile("global_load_async_to_lds_b128 %0, %1, off"
                 :: "v"(lds0), "v"(ga0) : "memory");
    if (lane < 15) {
        unsigned int       lds1 = (unsigned int)(size_t)&stage[buf][128 + lane * 4];
        unsigned long long ga1  = (unsigned long long)(size_t)(rowp + 128 + lane * 4);
        asm volatile("global_load_async_to_lds_b128 %0, %1, off"
                     :: "v"(lds1), "v"(ga1) : "memory");
    }
#else
    // host pass / fallback: direct copy (never executed on device path)
    const unsigned int* rowp = (const unsigned int*)(sup16 + (size_t)row * CWORDS16);
    for (int s = 0; s < 6; ++s) {
        int w = s * 32 + lane;
        if (w < CWORDS32) stage[buf][w] = rowp[w];
    }
#endif
}

__device__ __forceinline__ void wait_async_le2()
{
#if defined(__HIP_DEVICE_COMPILE__)
    asm volatile("s_wait_asynccnt 0x2" ::: "memory");
#endif
}

// ---------------- Kernel G: single-wave sequential suppression ----------------
// The dependent 6000-iteration chain is the latency-critical part of the whole
// pipeline. Double-buffered async global->LDS staging hides the 752B row fetch
// behind the previous iteration; s_wait_asynccnt<=2 both (a) proves the buffer
// being overwritten is idle (async loads complete in order) and (b) proves the
// current row has landed before the OR.
__global__ __launch_bounds__(32) void nms_seq_kernel(
    const unsigned short* __restrict__ sup16, const float* __restrict__ top_sc,
    unsigned int* __restrict__ keepWord, int* __restrict__ wpfx)
{
    __shared__ unsigned int remv[CWORDS32];
    __shared__ int cnts[CWORDS32];
    __shared__ unsigned int stage[2][CWORDS32];
    int lane = threadIdx.x;

    // init removed-mask: invalid (-inf) entries start removed; bits >= 6000 removed
#pragma unroll
    for (int s = 0; s < 6; ++s) {
        int w = s * 32 + lane;
        if (w < CWORDS32) {
            unsigned int r = 0;
            for (int b = 0; b < 32; ++b) {
                int j = w * 32 + b;
                bool inval = (j >= PRE_NMS) || !isfinite(top_sc[j]);
                r |= (inval ? 1u : 0u) << b;
            }
            remv[w] = r;
        }
    }
    __syncthreads();

    async_fetch_row(sup16, 0, stage, 0, lane);   // preload row 0

    for (int i = 0; i < PRE_NMS; ++i) {
        // buffer (i+1)&1 was last used for row i-1 (issued two iterations ago);
        // wait until only row i's fetch is outstanding, then prefetch row i+1.
        wait_async_le2();
        int nxt = (i + 1 < PRE_NMS) ? (i + 1) : (PRE_NMS - 1);
        async_fetch_row(sup16, nxt, stage, (i + 1) & 1, lane);

        unsigned int cur = remv[i >> 5];          // uniform across wave
        if (!((cur >> (i & 31)) & 1u)) {
            wait_async_le2();                     // row i is now in LDS
            const unsigned int* srow = stage[i & 1];
#pragma unroll
            for (int s = 0; s < 6; ++s) {
                int w = s * 32 + lane;
                if (w < CWORDS32) {
                    unsigned int v = srow[w];
                    if (w == CWORDS32 - 1) v &= 0xFFFFu;  // pad columns 6000..6015
                    remv[w] |= v;
                }
            }
        }
        __syncthreads();
    }

    // keep words + per-word popcounts
#pragma unroll
    for (int s = 0; s < 6; ++s) {
        int w = s * 32 + lane;
        if (w < CWORDS32) {
            unsigned int kw = ~remv[w];
            if (w == CWORDS32 - 1) kw &= 0xFFFFu;
            keepWord[w] = kw;
            cnts[w] = __popc(kw);
        }
    }
    __syncthreads();
    if (lane == 0) {
        int run = 0;
        for (int w = 0; w < CWORDS32; ++w) { wpfx[w] = run; run += cnts[w]; }
        wpfx[CWORDS32] = run;
    }
}

// ---------------- Kernel H: scatter survivors to output ----------------
__global__ __launch_bounds__(256) void write_kernel(
    const unsigned int* __restrict__ keepWord, const int* __restrict__ wpfx,
    const float* __restrict__ top_p, const float* __restrict__ top_sc,
    float* __restrict__ out)
{
    int j = blockIdx.x * 256 + threadIdx.x;
    if (j >= PRE_NMS) return;
    int w = j >> 5, b = j & 31;
    unsigned int kw = keepWord[w];
    if ((kw >> b) & 1u) {
        int rank = wpfx[w] + __popc(kw & ((1u << b) - 1u));
        if (rank < POST_NMS) {
            out[rank * 5 + 0] = 0.f;
            out[rank * 5 + 1] = top_p[j * 4 + 0];
            out[rank * 5 + 2] = top_p[j * 4 + 1];
            out[rank * 5 + 3] = top_p[j * 4 + 2];
            out[rank * 5 + 4] = top_p[j * 4 + 3];
            out[POST_NMS * 5 + rank] = top_sc[j];
        }
    }
}

// ---------------- workspace layout (bytes) ----------------
#define WS_PROPS    0u
#define WS_SC       589824u
#define WS_SUP      737280u
#define WS_TOPP     5249280u
#define WS_TOPSC    5345280u
#define WS_HIST     5369280u
#define WS_META     5377472u
#define WS_SELSC    5377536u
#define WS_SELIDX   5410304u
#define WS_KEEPW    5443072u
#define WS_WPFX     5443824u

extern "C" void kernel_launch(void* const* d_in, const int* in_sizes, int n_in,
                              void* d_out, int out_size, void* d_ws, size_t ws_size,
                              hipStream_t stream) {
    const float* deltas = (const float*)d_in[0];
    const float* scores = (const float*)d_in[1];
    const int*   imw    = (const int*)d_in[2];
    const int*   imh    = (const int*)d_in[3];
    float* out = (float*)d_out;
    char* ws = (char*)d_ws;

    float* props            = (float*)(ws + WS_PROPS);
    float* sc               = (float*)(ws + WS_SC);
    unsigned short* sup16   = (unsigned short*)(ws + WS_SUP);
    float* top_p            = (float*)(ws + WS_TOPP);
    float* top_sc           = (float*)(ws + WS_TOPSC);
    int*   hist             = (int*)(ws + WS_HIST);
    int*   meta             = (int*)(ws + WS_META);
    float* selScore         = (float*)(ws + WS_SELSC);
    int*   selIdx           = (int*)(ws + WS_SELIDX);
    unsigned int* keepWord  = (unsigned int*)(ws + WS_KEEPW);
    int*   wpfx             = (int*)(ws + WS_WPFX);

    // per-call init (graph-capture safe)
    hipMemsetAsync(hist, 0, NBINS * sizeof(int) + 64, stream);          // hist + meta
    hipMemsetAsync(out, 0, (size_t)out_size * sizeof(float), stream);   // blob + scores

    decode_kernel<<<(N_ALL + 255) / 256, 256, 0, stream>>>(deltas, scores, imw, imh, props, sc);
    hist_kernel<<<(N_ALL + 255) / 256, 256, 0, stream>>>(sc, hist);
    scan_kernel<<<1, 1, 0, stream>>>(hist, meta);
    select_kernel<<<(N_ALL + 255) / 256, 256, 0, stream>>>(sc, meta, selScore, selIdx);
    sort_kernel<<<1, 1024, 0, stream>>>(meta, selScore, selIdx, props, top_p, top_sc);
    sup_kernel<<<dim3(PRE_NMS / 16, PRE_NMS / 16), 32, 0, stream>>>(top_p, sup16);
    nms_seq_kernel<<<1, 32, 0, stream>>>(sup16, top_sc, keepWord, wpfx);
    write_kernel<<<(PRE_NMS + 255) / 256, 256, 0, stream>>>(keepWord, wpfx, top_p, top_sc, out);
}